// PathWeightedFFF_21466246545427
// MI455X (gfx1250) — compile-verified
//
#include <hip/hip_runtime.h>
#include <math.h>
#include <stdint.h>

typedef __attribute__((ext_vector_type(2))) float v2f;
typedef __attribute__((ext_vector_type(8))) float v8f;

#define T_TOK   2048
#define DIM     256
#define LVLS    8
#define NNODES  255
#define NPAIRS  (T_TOK * LVLS)      // 16384
#define MAXTILES 1280
#define AP      260                  // padded A row stride (floats): conflict-free b64 frag reads

// workspace layout (4-byte units)
#define WS_NODES  0                    // int[NPAIRS]
#define WS_PW     (WS_NODES + NPAIRS)  // float[NPAIRS]
#define WS_CNT    (WS_PW + NPAIRS)     // int[256]
#define WS_OFF    (WS_CNT + 256)       // int[256]
#define WS_CUR    (WS_OFF + 256)       // int[256]
#define WS_PTOK   (WS_CUR + 256)       // int[NPAIRS]
#define WS_PPW    (WS_PTOK + NPAIRS)   // float[NPAIRS]
#define WS_TNODE  (WS_PPW + NPAIRS)    // int[MAXTILES]
#define WS_TSTART (WS_TNODE + MAXTILES)// int[MAXTILES]
#define WS_NT     (WS_TSTART + MAXTILES)// int[1]

__global__ void k_zero(int* counts) {
    counts[threadIdx.x] = 0;
}

// One wave (32 lanes) per token: walk the tree, record node ids + GELU(|logit|),
// histogram node populations.
__global__ void k_route(const float* __restrict__ x, const float* __restrict__ w1s,
                        const float* __restrict__ b1s,
                        int* __restrict__ nodes, float* __restrict__ pw,
                        int* __restrict__ counts) {
    int wave = threadIdx.x >> 5;
    int lane = threadIdx.x & 31;
    int t = blockIdx.x * 4 + wave;
    const float* xr = x + t * DIM;
    float xv[8];
#pragma unroll
    for (int j = 0; j < 8; j++) xv[j] = xr[lane + 32 * j];
    int cur = 0;
#pragma unroll
    for (int i = 0; i < LVLS; i++) {
        const float* wr = w1s + cur * DIM;
        float p = 0.f;
#pragma unroll
        for (int j = 0; j < 8; j++) p += xv[j] * wr[lane + 32 * j];
#pragma unroll
        for (int off = 16; off > 0; off >>= 1) p += __shfl_xor(p, off, 32);
        float score = p + b1s[cur];
        float a = fabsf(score);
        float g = 0.5f * a * (1.0f + erff(a * 0.70710678118654752440f)); // exact-erf GELU
        if (lane == 0) {
            nodes[t * LVLS + i] = cur;
            pw[t * LVLS + i] = g;
            atomicAdd(&counts[cur], 1);
        }
        int choice = (score > 0.0f) ? 1 : 0;   // sign(0)->0 -> choice 0
        cur = cur * 2 + choice + 1;
    }
}

// out[t][o] = sum_i pw_i * b2s[node_i][o]   (initializes out; GEMM adds on top)
__global__ void k_bias(const float* __restrict__ b2s, const int* __restrict__ nodes,
                       const float* __restrict__ pw, float* __restrict__ out) {
    int t = blockIdx.x;
    int o = threadIdx.x;
    float acc = 0.f;
#pragma unroll
    for (int i = 0; i < LVLS; i++) {
        int n = nodes[t * LVLS + i];
        acc += pw[t * LVLS + i] * b2s[n * DIM + o];
    }
    out[t * DIM + o] = acc;
}

// Exclusive scan of counts + build tile descriptors (single block).
__global__ void k_scan(const int* __restrict__ counts, int* __restrict__ offsets,
                       int* __restrict__ cursors, int* __restrict__ tileNode,
                       int* __restrict__ tileStart, int* __restrict__ numTiles) {
    __shared__ int sc[256];
    int tid = threadIdx.x;
    int c = (tid < NNODES) ? counts[tid] : 0;
    sc[tid] = c;
    for (int off = 1; off < 256; off <<= 1) {
        __syncthreads();
        int v = (tid >= off) ? sc[tid - off] : 0;
        __syncthreads();
        sc[tid] += v;
    }
    __syncthreads();
    offsets[tid] = sc[tid] - c;   // exclusive
    cursors[tid] = 0;
    __syncthreads();
    if (tid == 0) {
        int nt = 0;
        for (int n = 0; n < NNODES; n++) {
            int cnt = counts[n];
            int off0 = sc[n] - cnt;
            for (int s = 0; s < cnt; s += 16) {
                tileNode[nt] = n;
                tileStart[nt] = off0 + s;
                nt++;
            }
        }
        *numTiles = nt;
    }
}

// Scatter (token, level) pairs into per-node buckets.
__global__ void k_scatter(const int* __restrict__ nodes, const float* __restrict__ pw,
                          const int* __restrict__ offsets, int* __restrict__ cursors,
                          int* __restrict__ ptok, float* __restrict__ ppw) {
    int g = blockIdx.x * 256 + threadIdx.x;   // 0..NPAIRS-1
    int n = nodes[g];
    int pos = atomicAdd(&cursors[n], 1);
    int idx = offsets[n] + pos;
    ptok[idx] = g >> 3;
    ppw[idx] = pw[g];
}

// Grouped GEMM: one block per 16-token tile of a node bucket.
// A (16x256) = raw gathered x rows, async-DMA'd into LDS (ASYNCcnt path),
//              row-major with AP=260 padded stride -> conflict-free ds_load_b64 frags.
// B (256x256) = w2s[node], streamed from global (L2-resident).
// 4 waves x 4 N-tiles of 16 cover N=256; K in steps of 4 via V_WMMA_F32_16X16X4_F32.
// Path weight pw is applied to the C rows at store time.
__global__ void k_gemm(const float* __restrict__ x, const float* __restrict__ w2s,
                       const int* __restrict__ ptok, const float* __restrict__ ppw,
                       const int* __restrict__ offsets, const int* __restrict__ counts,
                       const int* __restrict__ tileNode, const int* __restrict__ tileStart,
                       const int* __restrict__ numTiles, float* __restrict__ out) {
    int tile = blockIdx.x;
    if (tile >= *numTiles) return;
    int node  = tileNode[tile];
    int start = tileStart[tile];
    int end   = offsets[node] + counts[node];

    __shared__ float As[16 * AP];    // [m][k], padded stride AP floats (1040 B, 16B aligned)
    __shared__ int   toks[16];
    __shared__ float pws[16];

    int tid = threadIdx.x;
    if (tid < 16) {
        int p = start + tid;
        if (p < end) { toks[tid] = ptok[p]; pws[tid] = ppw[p]; }
        else         { toks[tid] = -1;      pws[tid] = 0.f;   }
    }
    __syncthreads();

    // Async gather of 16 x-rows into LDS: 1024 chunks of 16B, 8 per thread.
    uint32_t abase = (uint32_t)(uintptr_t)As;   // LDS byte address of As[0]
#pragma unroll
    for (int it = 0; it < 8; it++) {
        int c = tid + it * 128;        // 0..1023
        int m = c >> 6;                // row 0..15
        int q = c & 63;                // 16B chunk in row
        int tk = toks[m];
        int tok = (tk >= 0) ? tk : 0;  // padding rows gather row 0; results never stored
        const float* g = x + (size_t)tok * DIM + q * 4;
        uint32_t l = abase + (uint32_t)(m * (AP * 4) + q * 16);
        asm volatile("global_load_async_to_lds_b128 %0, %1, off"
                     :: "v"(l), "v"(g) : "memory");
    }
    asm volatile("s_wait_asynccnt 0x0" ::: "memory");
    __syncthreads();

    int wave = tid >> 5;
    int lane = tid & 31;
    int col0 = wave * 64;
    int mlo  = lane & 15;
    int khalf = (lane >> 4) * 2;   // lanes 0-15 -> K 0,1 ; lanes 16-31 -> K 2,3
    const float* W = w2s + (size_t)node * (DIM * DIM);

    v8f c0 = {}, c1 = {}, c2 = {}, c3 = {};

    for (int kb = 0; kb < DIM; kb += 4) {
        int ks = kb + khalf;
        v2f a = *(const v2f*)&As[mlo * AP + ks];   // one ds_load_b64, conflict-free
        const float* Wk = W + (size_t)ks * DIM;
        int col = col0 + mlo;
        v2f b;
        b.x = Wk[col];        b.y = Wk[DIM + col];
        c0 = __builtin_amdgcn_wmma_f32_16x16x4_f32(false, a, false, b, (short)0, c0, false, false);
        b.x = Wk[col + 16];   b.y = Wk[DIM + col + 16];
        c1 = __builtin_amdgcn_wmma_f32_16x16x4_f32(false, a, false, b, (short)0, c1, false, false);
        b.x = Wk[col + 32];   b.y = Wk[DIM + col + 32];
        c2 = __builtin_amdgcn_wmma_f32_16x16x4_f32(false, a, false, b, (short)0, c2, false, false);
        b.x = Wk[col + 48];   b.y = Wk[DIM + col + 48];
        c3 = __builtin_amdgcn_wmma_f32_16x16x4_f32(false, a, false, b, (short)0, c3, false, false);
    }

#pragma unroll
    for (int r = 0; r < 8; r++) {
        int m = r + ((lane >> 4) << 3);   // VGPR r: lanes 0-15 -> M=r, lanes 16-31 -> M=r+8
        int tk = toks[m];
        if (tk >= 0) {
            float w = pws[m];
            float* orow = out + (size_t)tk * DIM;
            int cb = col0 + mlo;
            atomicAdd(orow + cb,      w * c0[r]);
            atomicAdd(orow + cb + 16, w * c1[r]);
            atomicAdd(orow + cb + 32, w * c2[r]);
            atomicAdd(orow + cb + 48, w * c3[r]);
        }
    }
}

extern "C" void kernel_launch(void* const* d_in, const int* in_sizes, int n_in,
                              void* d_out, int out_size, void* d_ws, size_t ws_size,
                              hipStream_t stream) {
    const float* x   = (const float*)d_in[0];
    const float* w1s = (const float*)d_in[1];
    const float* b1s = (const float*)d_in[2];
    const float* w2s = (const float*)d_in[3];
    const float* b2s = (const float*)d_in[4];
    float* out = (float*)d_out;

    int*   wsI = (int*)d_ws;
    float* wsF = (float*)d_ws;
    int*   nodes    = wsI + WS_NODES;
    float* pw       = wsF + WS_PW;
    int*   counts   = wsI + WS_CNT;
    int*   offsets  = wsI + WS_OFF;
    int*   cursors  = wsI + WS_CUR;
    int*   ptok     = wsI + WS_PTOK;
    float* ppw      = wsF + WS_PPW;
    int*   tileNode = wsI + WS_TNODE;
    int*   tileStart= wsI + WS_TSTART;
    int*   numTiles = wsI + WS_NT;

    k_zero<<<1, 256, 0, stream>>>(counts);
    k_route<<<T_TOK / 4, 128, 0, stream>>>(x, w1s, b1s, nodes, pw, counts);
    k_bias<<<T_TOK, 256, 0, stream>>>(b2s, nodes, pw, out);
    k_scan<<<1, 256, 0, stream>>>(counts, offsets, cursors, tileNode, tileStart, numTiles);
    k_scatter<<<NPAIRS / 256, 256, 0, stream>>>(nodes, pw, offsets, cursors, ptok, ppw);
    k_gemm<<<MAXTILES, 128, 0, stream>>>(x, w2s, ptok, ppw, offsets, counts,
                                         tileNode, tileStart, numTiles, out);
}